// ProteinEncoderEGNN_46523085750931
// MI455X (gfx1250) — compile-verified
//
#include <hip/hip_runtime.h>
#include <math.h>

// -------------------------------------------------------------------------
// EGNN protein encoder for MI455X (gfx1250, wave32).
//
// Edge message GEMM (E x 2577 @ 2577 x 24) is hoisted algebraically into a
// per-node projection GEMM P = h @ [msgW1a | msgW1b | nodeW1a] (N x 1280 @
// 1280 x 72). Dense GEMMs run on V_WMMA_F32_16X16X4_F32 (full fp32).
// Staging: A panel via double-buffered GLOBAL_LOAD_ASYNC_TO_LDS_B128
// (ASYNCcnt-tracked DMA overlapping WMMA); B panel via pipelined global
// float4 load -> transposed LDS store so B fragments are contiguous b64
// reads feeding WMMA operand pairs directly (no v_mov shuffles).
// -------------------------------------------------------------------------

#define N_NODES 10000
#define N_EDGES 160000
#define DIM     1280
#define MDIM    24
#define EEDIM   16
#define LAYERS  6
#define PDIM    72     // 24 (msg W1a) + 24 (msg W1b) + 24 (node W1a)

typedef __attribute__((ext_vector_type(2))) float v2f;
typedef __attribute__((ext_vector_type(8))) float v8f;

__device__ __forceinline__ float silu_f(float x) {
    return x / (1.0f + __expf(-x));
}

// Async global -> LDS copy of 16 bytes (CDNA5 GLOBAL_LOAD_ASYNC_TO_LDS_B128,
// tracked by ASYNCcnt). LDS byte offset = low 32 bits of the generic pointer
__device__ __forceinline__ void async_b128(void* lds_ptr, const float* gptr) {
    unsigned lds_off = (unsigned)(size_t)lds_ptr;
    unsigned long long ga = (unsigned long long)(size_t)gptr;
    asm volatile("global_load_async_to_lds_b128 %0, %1, off"
                 :: "v"(lds_off), "v"(ga) : "memory");
}

__device__ __forceinline__ void wait_async0() {
    asm volatile("s_wait_asynccnt 0x0" ::: "memory");
}

// -------------------------------------------------------------------------
// fp32 WMMA GEMM: C(nrows x ncols) = A(nrows x K) @ B(K x ncols) [+ bias]
// Block = 256 threads = 8 waves; tile = 128 rows x 64 cols; K staged 16-wide
// through double-buffered LDS. Requires K % 16 == 0 and ncols % 4 == 0
// (true here: K = 1280, ncols in {1280, 72}).
// -------------------------------------------------------------------------
__global__ __launch_bounds__(256)
void wmma_gemm_f32(const float* __restrict__ A, int lda,
                   const float* __restrict__ B, int ldb,
                   const float* __restrict__ bias,
                   float* __restrict__ C, int ldc,
                   int nrows, int ncols, int K)
{
    // stride 20 floats (80B): async b128 writes stay 16B aligned; fragment
    // reads (16 lanes, row stride 20) hit 16 distinct banks; b64 reads are
    // 8B aligned (row*20 + even col).
    __shared__ float As[2][128][20];   // [row][k]
    __shared__ float Bs[2][64][20];    // transposed [n][k]

    const int t    = threadIdx.x;
    const int lane = t & 31;
    const int wv   = t >> 5;        // wave id 0..7 -> 16-row strip
    const int half = lane >> 4;     // 0: K+0..1, 1: K+2..3 (ISA A/B layout)
    const int r    = lane & 15;
    const int m0   = blockIdx.y * 128;
    const int n0   = blockIdx.x * 64;

    // Zero both buffers once. Staging only overwrites in-bounds entries,
    // so guard-skipped rows/cols read as zero for the whole K loop.
    for (int i = t; i < 2 * 128 * 20; i += 256) ((float*)As)[i] = 0.f;
    for (int i = t; i < 2 * 64 * 20;  i += 256) ((float*)Bs)[i] = 0.f;
    __syncthreads();

    // per-thread staging coordinates
    const int ar   = t >> 2;            // 0..63
    const int ar2  = ar + 64;           // 64..127
    const int aseg = (t & 3) * 4;       // 0,4,8,12
    const int brow = t >> 4;            // 0..15 (k)
    const int bseg = (t & 15) * 4;      // 0..60 (n)
    const bool aok  = (m0 + ar)  < nrows;
    const bool aok2 = (m0 + ar2) < nrows;
    const bool bok  = (n0 + bseg + 4) <= ncols;

    v8f acc0 = {}; v8f acc1 = {}; v8f acc2 = {}; v8f acc3 = {};

    // prologue: stage panel 0 into buffer 0
    if (aok)  async_b128(&As[0][ar][aseg],  A + (size_t)(m0 + ar)  * lda + aseg);
    if (aok2) async_b128(&As[0][ar2][aseg], A + (size_t)(m0 + ar2) * lda + aseg);
    if (bok) {
        float4 bv = *(const float4*)(B + (size_t)brow * ldb + n0 + bseg);
        Bs[0][bseg + 0][brow] = bv.x; Bs[0][bseg + 1][brow] = bv.y;
        Bs[0][bseg + 2][brow] = bv.z; Bs[0][bseg + 3][brow] = bv.w;
    }

    for (int k0 = 0; k0 < K; k0 += 16) {
        const int buf = (k0 >> 4) & 1;
        wait_async0();          // this wave's A-panel DMA complete
        __syncthreads();        // all waves' A DMA + B stores visible

        // kick off next panel (overlaps compute): async A DMA into buf^1,
        // B global load into registers (stored to LDS after compute).
        float4 bv = make_float4(0.f, 0.f, 0.f, 0.f);
        const bool more = (k0 + 16) < K;
        if (more) {
            const int kn = k0 + 16;
            if (aok)  async_b128(&As[buf ^ 1][ar][aseg],
                                 A + (size_t)(m0 + ar)  * lda + kn + aseg);
            if (aok2) async_b128(&As[buf ^ 1][ar2][aseg],
                                 A + (size_t)(m0 + ar2) * lda + kn + aseg);
            if (bok)  bv = *(const float4*)(B + (size_t)(kn + brow) * ldb + n0 + bseg);
        }

        #pragma unroll
        for (int kk = 0; kk < 16; kk += 4) {
            int c = kk + 2 * half;
            v2f a;  a.x  = As[buf][wv * 16 + r][c]; a.y  = As[buf][wv * 16 + r][c + 1];
            v2f b0; b0.x = Bs[buf][ 0 + r][c];      b0.y = Bs[buf][ 0 + r][c + 1];
            v2f b1; b1.x = Bs[buf][16 + r][c];      b1.y = Bs[buf][16 + r][c + 1];
            v2f b2; b2.x = Bs[buf][32 + r][c];      b2.y = Bs[buf][32 + r][c + 1];
            v2f b3; b3.x = Bs[buf][48 + r][c];      b3.y = Bs[buf][48 + r][c + 1];
            acc0 = __builtin_amdgcn_wmma_f32_16x16x4_f32(false, a, false, b0, (short)0, acc0, false, false);
            acc1 = __builtin_amdgcn_wmma_f32_16x16x4_f32(false, a, false, b1, (short)0, acc1, false, false);
            acc2 = __builtin_amdgcn_wmma_f32_16x16x4_f32(false, a, false, b2, (short)0, acc2, false, false);
            acc3 = __builtin_amdgcn_wmma_f32_16x16x4_f32(false, a, false, b3, (short)0, acc3, false, false);
        }

        // transpose-store the pipelined B panel into buf^1; the next
        // iteration's barrier publishes it (all reads of buf^1 finished
        // before this iteration's barrier).
        if (more && bok) {
            Bs[buf ^ 1][bseg + 0][brow] = bv.x;
            Bs[buf ^ 1][bseg + 1][brow] = bv.y;
            Bs[buf ^ 1][bseg + 2][brow] = bv.z;
            Bs[buf ^ 1][bseg + 3][brow] = bv.w;
        }
    }

    // ---- store (ISA C layout: row = v + 8*half, col = lane&15) ----
    int rbase = m0 + wv * 16 + half * 8;
    #pragma unroll
    for (int nt = 0; nt < 4; ++nt) {
        v8f av = (nt == 0) ? acc0 : (nt == 1) ? acc1 : (nt == 2) ? acc2 : acc3;
        int col = n0 + nt * 16 + r;
        if (col >= ncols) continue;
        float bb = bias ? bias[col] : 0.f;
        #pragma unroll
        for (int v = 0; v < 8; ++v) {
            int row = rbase + v;
            if (row < nrows) C[(size_t)row * ldc + col] = av[v] + bb;
        }
    }
}

// -------------------------------------------------------------------------
// Pack per-layer projection matrix Bp[l] (1280 x 72) =
//   [msg_W1[l][0:1280] | msg_W1[l][1280:2560] | node_W1[l][0:1280]]
// -------------------------------------------------------------------------
__global__ void pack_weights_kernel(const float* __restrict__ msgW1,
                                    const float* __restrict__ nodeW1,
                                    float* __restrict__ Bp)
{
    int i = blockIdx.x * blockDim.x + threadIdx.x;
    if (i >= LAYERS * DIM * PDIM) return;
    int c = i % PDIM;
    int k = (i / PDIM) % DIM;
    int l = i / (PDIM * DIM);
    float v;
    if (c < 24)
        v = msgW1[(size_t)l * 2577 * 24 + (size_t)k * 24 + c];
    else if (c < 48)
        v = msgW1[(size_t)l * 2577 * 24 + (size_t)(1280 + k) * 24 + (c - 24)];
    else
        v = nodeW1[(size_t)l * 1304 * 24 + (size_t)k * 24 + (c - 48)];
    Bp[i] = v;
}

// zero degree counters + copy initial positions into working buffer
__global__ void init_kernel(const float* __restrict__ pos_in,
                            float* __restrict__ pos_a,
                            float* __restrict__ deg)
{
    int i = blockIdx.x * blockDim.x + threadIdx.x;
    if (i < N_NODES) deg[i] = 0.f;
    if (i < N_NODES * 3) pos_a[i] = pos_in[i];
}

// sinusoidal distance embedding (from initial pos) + in-degree of col
__global__ void edge_init_kernel(const float* __restrict__ pos,
                                 const int* __restrict__ eidx,
                                 float* __restrict__ edge_attr,
                                 float* __restrict__ deg)
{
    int e = blockIdx.x * blockDim.x + threadIdx.x;
    if (e >= N_EDGES) return;
    int rw = eidx[e];
    int cl = eidx[N_EDGES + e];
    float dx = pos[rw * 3 + 0] - pos[cl * 3 + 0];
    float dy = pos[rw * 3 + 1] - pos[cl * 3 + 1];
    float dz = pos[rw * 3 + 2] - pos[cl * 3 + 2];
    float dist = sqrtf(dx * dx + dy * dy + dz * dz);
    #pragma unroll
    for (int i = 0; i < 8; ++i) {
        // freq_i = exp(-ln(10000) * 2i / 16) = exp(-1.15129254649702 * i)
        float freq = __expf(-1.15129254649702f * (float)i);
        float ang = dist * freq;
        edge_attr[(size_t)e * EEDIM + 2 * i + 0] = __sinf(ang);
        edge_attr[(size_t)e * EEDIM + 2 * i + 1] = __cosf(ang);
    }
    atomicAdd(&deg[cl], 1.0f);
}

// per-layer: zero msg_agg; seed pos_next = pos_cur for coord atomics
__global__ void prep_layer_kernel(float* __restrict__ msg_agg,
                                  const float* __restrict__ pos_src,
                                  float* __restrict__ pos_dst, int do_pos)
{
    int i = blockIdx.x * blockDim.x + threadIdx.x;
    if (i < N_NODES * MDIM) msg_agg[i] = 0.f;
    if (do_pos && i < N_NODES * 3) pos_dst[i] = pos_src[i];
}

// -------------------------------------------------------------------------
// Per-edge message MLP (after algebraic hoist), scatter into msg_agg and
// (layers 0..4) coordinate update. One edge per lane; weights in LDS.
// -------------------------------------------------------------------------
__global__ __launch_bounds__(256)
void edge_msg_kernel(const int* __restrict__ eidx,
                     const float* __restrict__ pos_cur,
                     float* __restrict__ pos_nxt,
                     const float* __restrict__ P,
                     const float* __restrict__ edge_attr,
                     const float* __restrict__ W1c,   // 16 x 24
                     const float* __restrict__ W1d,   // 24 (radial row)
                     const float* __restrict__ b1,    // 24
                     const float* __restrict__ W2,    // 24 x 24
                     const float* __restrict__ b2,    // 24
                     const float* __restrict__ cW1,   // 24 x 24
                     const float* __restrict__ cb1,   // 24
                     const float* __restrict__ cW2,   // 24
                     float* __restrict__ msg_agg,
                     int update_pos)
{
    __shared__ float sW1c[16 * 24], sW2[24 * 24], scW1[24 * 24];
    __shared__ float sW1d[24], sb1[24], sb2[24], scb1[24], scW2[24];

    for (int i = threadIdx.x; i < 16 * 24; i += 256) sW1c[i] = W1c[i];
    for (int i = threadIdx.x; i < 24 * 24; i += 256) sW2[i] = W2[i];
    if (threadIdx.x < 24) {
        int m = threadIdx.x;
        sW1d[m] = W1d[m]; sb1[m] = b1[m]; sb2[m] = b2[m];
    }
    if (update_pos) {
        for (int i = threadIdx.x; i < 24 * 24; i += 256) scW1[i] = cW1[i];
        if (threadIdx.x < 24) {
            int m = threadIdx.x;
            scb1[m] = cb1[m]; scW2[m] = cW2[m];
        }
    }
    __syncthreads();

    int e = blockIdx.x * blockDim.x + threadIdx.x;
    if (e >= N_EDGES) return;
    int rw = eidx[e];            // source j
    int cl = eidx[N_EDGES + e];  // target i
    float cdx = pos_cur[rw * 3 + 0] - pos_cur[cl * 3 + 0];
    float cdy = pos_cur[rw * 3 + 1] - pos_cur[cl * 3 + 1];
    float cdz = pos_cur[rw * 3 + 2] - pos_cur[cl * 3 + 2];
    float radial = cdx * cdx + cdy * cdy + cdz * cdz;

    const float* Pa = P + (size_t)cl * PDIM;        // h[col] @ W1a
    const float* Pb = P + (size_t)rw * PDIM + 24;   // h[row] @ W1b

    float pre[24];
    #pragma unroll
    for (int m = 0; m < 24; ++m)
        pre[m] = Pa[m] + Pb[m] + radial * sW1d[m] + sb1[m];
    #pragma unroll
    for (int j = 0; j < 16; ++j) {
        float ea = edge_attr[(size_t)e * EEDIM + j];
        #pragma unroll
        for (int m = 0; m < 24; ++m) pre[m] += ea * sW1c[j * 24 + m];
    }
    #pragma unroll
    for (int m = 0; m < 24; ++m) pre[m] = silu_f(pre[m]);

    float msg[24];
    #pragma unroll
    for (int m = 0; m < 24; ++m) {
        float s = sb2[m];
        #pragma unroll
        for (int j = 0; j < 24; ++j) s += pre[j] * sW2[j * 24 + m];
        msg[m] = silu_f(s);
    }

    float* agg = msg_agg + (size_t)cl * MDIM;
    #pragma unroll
    for (int m = 0; m < 24; ++m) atomicAdd(&agg[m], msg[m]);

    if (update_pos) {
        float w = 0.f;
        #pragma unroll
        for (int m = 0; m < 24; ++m) {
            float s = scb1[m];
            #pragma unroll
            for (int j = 0; j < 24; ++j) s += msg[j] * scW1[j * 24 + m];
            w += silu_f(s) * scW2[m];
        }
        atomicAdd(&pos_nxt[cl * 3 + 0], cdx * w);
        atomicAdd(&pos_nxt[cl * 3 + 1], cdy * w);
        atomicAdd(&pos_nxt[cl * 3 + 2], cdz * w);
    }
}

// -------------------------------------------------------------------------
// Node update + residual + LayerNorm. One block (256 thr) per node.
// h @ nodeW1a already in P[:,48:72]; only the 24x24 msg_agg part remains.
// -------------------------------------------------------------------------
__global__ __launch_bounds__(256)
void node_update_kernel(float* __restrict__ h,
                        const float* __restrict__ P,
                        const float* __restrict__ msg_agg,
                        const float* __restrict__ deg,
                        const float* __restrict__ nW1,  // + l*1304*24
                        const float* __restrict__ nb1,
                        const float* __restrict__ nW2,  // 24 x 1280
                        const float* __restrict__ nb2,
                        const float* __restrict__ g,
                        const float* __restrict__ bln)
{
    __shared__ float mm[24], tt[24], sW1b[24 * 24], red[256];
    int n = blockIdx.x;
    int t = threadIdx.x;

    for (int i = t; i < 24 * 24; i += 256) {
        int j = i / 24, m = i % 24;
        sW1b[i] = nW1[(size_t)(1280 + j) * 24 + m];
    }
    if (t < 24) {
        float den = fmaxf(deg[n], 1.0f);
        mm[t] = msg_agg[(size_t)n * MDIM + t] / den;
    }
    __syncthreads();
    if (t < 24) {
        float s = P[(size_t)n * PDIM + 48 + t] + nb1[t];
        #pragma unroll
        for (int j = 0; j < 24; ++j) s += mm[j] * sW1b[j * 24 + t];
        tt[t] = silu_f(s);
    }
    __syncthreads();

    float hn[5];
    float lsum = 0.f, lsq = 0.f;
    #pragma unroll
    for (int i = 0; i < 5; ++i) {
        int c = t + 256 * i;
        float up = nb2[c];
        #pragma unroll
        for (int m = 0; m < 24; ++m) up += tt[m] * nW2[(size_t)m * DIM + c];
        float v = h[(size_t)n * DIM + c] + up;
        hn[i] = v; lsum += v; lsq += v * v;
    }
    red[t] = lsum; __syncthreads();
    for (int s = 128; s > 0; s >>= 1) {
        if (t < s) red[t] += red[t + s];
        __syncthreads();
    }
    float mu = red[0] * (1.f / (float)DIM);
    __syncthreads();
    red[t] = lsq; __syncthreads();
    for (int s = 128; s > 0; s >>= 1) {
        if (t < s) red[t] += red[t + s];
        __syncthreads();
    }
    float var  = red[0] * (1.f / (float)DIM) - mu * mu;
    float rstd = rsqrtf(var + 1e-5f);
    #pragma unroll
    for (int i = 0; i < 5; ++i) {
        int c = t + 256 * i;
        h[(size_t)n * DIM + c] = (hn[i] - mu) * rstd * g[c] + bln[c];
    }
}

__global__ void vcopy_kernel(const float* __restrict__ src,
                             float* __restrict__ dst, int n)
{
    int i = blockIdx.x * blockDim.x + threadIdx.x;
    if (i < n) dst[i] = src[i];
}

// -------------------------------------------------------------------------
extern "C" void kernel_launch(void* const* d_in, const int* in_sizes, int n_in,
                              void* d_out, int out_size, void* d_ws, size_t ws_size,
                              hipStream_t stream)
{
    const float* h_in     = (const float*)d_in[0];
    const float* pos_in   = (const float*)d_in[1];
    const int*   eidx     = (const int*)  d_in[2];
    const float* emb_W    = (const float*)d_in[3];
    const float* emb_b    = (const float*)d_in[4];
    const float* msg_W1   = (const float*)d_in[5];
    const float* msg_b1   = (const float*)d_in[6];
    const float* msg_W2   = (const float*)d_in[7];
    const float* msg_b2   = (const float*)d_in[8];
    const float* coord_W1 = (const float*)d_in[9];
    const float* coord_b1 = (const float*)d_in[10];
    const float* coord_W2 = (const float*)d_in[11];
    const float* node_W1  = (const float*)d_in[12];
    const float* node_b1  = (const float*)d_in[13];
    const float* node_W2  = (const float*)d_in[14];
    const float* node_b2  = (const float*)d_in[15];
    const float* ln_g     = (const float*)d_in[16];
    const float* ln_b     = (const float*)d_in[17];

    // workspace layout (~68 MB of fp32)
    float* ws = (float*)d_ws;
    size_t off = 0;
    float* h      = ws + off; off += (size_t)N_NODES * DIM;
    float* pos_a  = ws + off; off += (size_t)N_NODES * 3;
    float* pos_b  = ws + off; off += (size_t)N_NODES * 3;
    float* eattr  = ws + off; off += (size_t)N_EDGES * EEDIM;
    float* deg    = ws + off; off += (size_t)N_NODES;
    float* P      = ws + off; off += (size_t)N_NODES * PDIM;
    float* magg   = ws + off; off += (size_t)N_NODES * MDIM;
    float* Bp     = ws + off; off += (size_t)LAYERS * DIM * PDIM;

    {
        int tot = LAYERS * DIM * PDIM;
        pack_weights_kernel<<<(tot + 255) / 256, 256, 0, stream>>>(msg_W1, node_W1, Bp);
    }
    init_kernel<<<(N_NODES * 3 + 255) / 256, 256, 0, stream>>>(pos_in, pos_a, deg);
    edge_init_kernel<<<(N_EDGES + 255) / 256, 256, 0, stream>>>(pos_in, eidx, eattr, deg);

    // node embedding: h = h_in @ emb_W + emb_b   (10000x1280 @ 1280x1280)
    {
        dim3 grid((DIM + 63) / 64, (N_NODES + 127) / 128);
        wmma_gemm_f32<<<grid, 256, 0, stream>>>(h_in, DIM, emb_W, DIM, emb_b,
                                                h, DIM, N_NODES, DIM, DIM);
    }

    float* pos_cur = pos_a;
    float* pos_nxt = pos_b;
    for (int l = 0; l < LAYERS; ++l) {
        int upd = (l < LAYERS - 1) ? 1 : 0;

        // per-node projection P = h @ Bp[l]   (10000x1280 @ 1280x72)
        dim3 gridP((PDIM + 63) / 64, (N_NODES + 127) / 128);
        wmma_gemm_f32<<<gridP, 256, 0, stream>>>(h, DIM, Bp + (size_t)l * DIM * PDIM, PDIM,
                                                 nullptr, P, PDIM, N_NODES, PDIM, DIM);

        prep_layer_kernel<<<(N_NODES * MDIM + 255) / 256, 256, 0, stream>>>(
            magg, pos_cur, pos_nxt, upd);

        edge_msg_kernel<<<(N_EDGES + 255) / 256, 256, 0, stream>>>(
            eidx, pos_cur, pos_nxt, P, eattr,
            msg_W1 + (size_t)l * 2577 * 24 + (size_t)2560 * 24,  // W1c (edge_attr rows)
            msg_W1 + (size_t)l * 2577 * 24 + (size_t)2576 * 24,  // W1d (radial row)
            msg_b1 + l * 24,
            msg_W2 + (size_t)l * 576, msg_b2 + l * 24,
            upd ? (coord_W1 + (size_t)l * 576) : coord_W1,
            upd ? (coord_b1 + l * 24) : coord_b1,
            upd ? (coord_W2 + l * 24) : coord_W2,
            magg, upd);

        node_update_kernel<<<N_NODES, 256, 0, stream>>>(
            h, P, magg, deg,
            node_W1 + (size_t)l * 1304 * 24,
            node_b1 + l * 24,
            node_W2 + (size_t)l * 24 * DIM,
            node_b2 + (size_t)l * DIM,
            ln_g + (size_t)l * DIM, ln_b + (size_t)l * DIM);

        if (upd) { float* tmp = pos_cur; pos_cur = pos_nxt; pos_nxt = tmp; }
    }

    // outputs: h then pos, concatenated flat
    vcopy_kernel<<<((N_NODES * DIM) + 255) / 256, 256, 0, stream>>>(
        h, (float*)d_out, N_NODES * DIM);
    vcopy_kernel<<<(N_NODES * 3 + 255) / 256, 256, 0, stream>>>(
        pos_cur, (float*)d_out + (size_t)N_NODES * DIM, N_NODES * 3);
}